// AttnAggregator2_20529943675142
// MI455X (gfx1250) — compile-verified
//
#include <hip/hip_runtime.h>
#include <hip/hip_bf16.h>

// Problem constants (match reference): D=128, S=25 neighbors, rows = S+1 = 26.
#define DIM     128
#define NSAMP   25
#define ROWS    26          // self + 25 neighbors
#define MPAD    32          // rows padded to 2 WMMA M-tiles
#define PITCH   132         // row pitch (dwords) in LDS cat tile: 132%64==4 -> conflict-free b64 frag loads
#define KSTEPS  32          // 128 / 4 (f32 WMMA K=4)
#define NTILES  8           // 128 / 16 column tiles, one per wave
#define WGSIZE  256         // 8 wave32 waves

typedef __attribute__((ext_vector_type(2))) float v2f;
typedef __attribute__((ext_vector_type(8))) float v8f;

__launch_bounds__(WGSIZE, 1)
__global__ void attn_agg_kernel(const float* __restrict__ table,
                                const int*   __restrict__ node,
                                const int*   __restrict__ neighs,
                                const float* __restrict__ Wq, const float* __restrict__ bq,
                                const float* __restrict__ Wk, const float* __restrict__ bk,
                                const float* __restrict__ Wv, const float* __restrict__ bv,
                                float* __restrict__ out, int nNodes)
{
    // CDNA5: 320KB LDS per WGP lets us keep all three weight matrices resident
    // in WMMA B-fragment order (f32), amortized over ~N/gridDim nodes.
    __shared__ float2 wfrag[3 * KSTEPS * NTILES * 32];  // 192 KB: [mat][kstep][ntile][lane] = {W[n][k], W[n][k+1]}
    __shared__ float  cat[MPAD * PITCH];                // gathered (self+neigh) rows, padded
    __shared__ float  bias[3 * DIM];                    // bq | bk | bv
    __shared__ float  spart[NTILES * 32];               // per-wave partial scores
    __shared__ float  scores[32];                       // reduced scores (26 valid)

    const int tid  = threadIdx.x;
    const int wave = tid >> 5;
    const int lane = tid & 31;
    const int mrow = lane & 15;       // row within M-tile / column within N-tile
    const int half = lane >> 4;       // lane half selects K sub-pair (WMMA f32 layout)
    const int koff = half << 1;       // 0 or 2
    const int ncol = wave * 16 + mrow;

    // ---- one-time staging: weights (transposed, fragment order) + biases + zero pad rows ----
    {
        const float* Ws[3] = {Wq, Wk, Wv};
        for (int idx = tid; idx < 3 * KSTEPS * NTILES * 32; idx += WGSIZE) {
            int l  = idx & 31;
            int nt = (idx >> 5) & 7;
            int s  = (idx >> 8) & 31;
            int m  = idx >> 13;
            int n  = nt * 16 + (l & 15);
            int k  = 4 * s + ((l >> 4) << 1);
            wfrag[idx] = *(const float2*)(Ws[m] + n * DIM + k);   // nn.Linear: y = x @ W^T, so B[k][n] = W[n][k]
        }
        for (int idx = tid; idx < 3 * DIM; idx += WGSIZE) {
            const float* B = (idx < DIM) ? bq : ((idx < 2 * DIM) ? bk : bv);
            bias[idx] = B[idx & (DIM - 1)];
        }
        for (int idx = tid; idx < (MPAD - ROWS) * PITCH; idx += WGSIZE)
            cat[ROWS * PITCH + idx] = 0.0f;                        // pad rows 26..31 stay zero
    }
    __syncthreads();

    for (int nd = blockIdx.x; nd < nNodes; nd += gridDim.x) {
        // ---- phase A: gather 26 table rows into LDS (float4 segments) ----
        for (int idx = tid; idx < ROWS * 32; idx += WGSIZE) {
            int row = idx >> 5;
            int seg = idx & 31;
            int g = (row == 0) ? node[nd] : neighs[nd * NSAMP + (row - 1)];
            float4 v = *((const float4*)(table + (size_t)g * DIM) + seg);
            *(float4*)(cat + row * PITCH + seg * 4) = v;
        }
        // prefetch next node's rows into cache (global_prefetch_b8)
        {
            int nn = nd + gridDim.x;
            if (nn < nNodes && tid < ROWS) {
                int g = (tid == 0) ? node[nn] : neighs[nn * NSAMP + (tid - 1)];
                __builtin_prefetch(table + (size_t)g * DIM, 0, 0);
            }
        }
        __syncthreads();

        // ---- phase B: WMMA projections. Wave w owns output columns [16w, 16w+16). ----
        v8f accq  = {0.f,0.f,0.f,0.f,0.f,0.f,0.f,0.f};
        v8f acck0 = {0.f,0.f,0.f,0.f,0.f,0.f,0.f,0.f};
        v8f acck1 = {0.f,0.f,0.f,0.f,0.f,0.f,0.f,0.f};
        v8f accv0 = {0.f,0.f,0.f,0.f,0.f,0.f,0.f,0.f};
        v8f accv1 = {0.f,0.f,0.f,0.f,0.f,0.f,0.f,0.f};

        #pragma unroll 4
        for (int s = 0; s < KSTEPS; ++s) {
            // A fragments (f32 16x4): lane -> row mrow, VGPR pair = {A[m][k], A[m][k+1]}, k = 4s + 2*half
            v2f a0 = *(const v2f*)(cat + mrow * PITCH + 4 * s + koff);          // rows 0..15
            v2f a1 = *(const v2f*)(cat + (16 + mrow) * PITCH + 4 * s + koff);   // rows 16..31 (26.. zero)
            int bidx = (s * NTILES + wave) * 32 + lane;
            v2f bfq = *(const v2f*)&wfrag[bidx];
            v2f bfk = *(const v2f*)&wfrag[KSTEPS * NTILES * 32 + bidx];
            v2f bfv = *(const v2f*)&wfrag[2 * KSTEPS * NTILES * 32 + bidx];

            accq  = __builtin_amdgcn_wmma_f32_16x16x4_f32(false, a0, false, bfq, (short)0, accq,  false, false);
            acck0 = __builtin_amdgcn_wmma_f32_16x16x4_f32(false, a0, false, bfk, (short)0, acck0, false, false);
            acck1 = __builtin_amdgcn_wmma_f32_16x16x4_f32(false, a1, false, bfk, (short)0, acck1, false, false);
            accv0 = __builtin_amdgcn_wmma_f32_16x16x4_f32(false, a0, false, bfv, (short)0, accv0, false, false);
            accv1 = __builtin_amdgcn_wmma_f32_16x16x4_f32(false, a1, false, bfv, (short)0, accv1, false, false);
        }

        // biases (C layout: VGPR r -> M = r + 8*half, N = ncol)
        const float bkv = bias[DIM + ncol];
        const float bvv = bias[2 * DIM + ncol];
        #pragma unroll
        for (int r = 0; r < 8; ++r) {
            acck0[r] += bkv; acck1[r] += bkv;
            accv0[r] += bvv; accv1[r] += bvv;
        }

        // Q = row 0 of accq (lanes 0..15); broadcast to lanes 16..31
        float qv = accq[0] + bias[ncol];
        qv = __int_as_float(__builtin_amdgcn_ds_bpermute(mrow << 2, __float_as_int(qv)));

        // partial scores over this wave's 16 columns: reduce across the 16 lanes of each half
        float p0[8], p1[8];
        #pragma unroll
        for (int r = 0; r < 8; ++r) { p0[r] = qv * acck0[r]; p1[r] = qv * acck1[r]; }
        #pragma unroll
        for (int m = 1; m <= 8; m <<= 1) {
            #pragma unroll
            for (int r = 0; r < 8; ++r) {
                p0[r] += __shfl_xor(p0[r], m, 32);
                p1[r] += __shfl_xor(p1[r], m, 32);
            }
        }
        if (mrow == 0) {  // lanes 0 and 16 publish rows (r + 8*half) and (16 + r + 8*half)
            #pragma unroll
            for (int r = 0; r < 8; ++r) {
                spart[wave * 32 + half * 8 + r]      = p0[r];
                spart[wave * 32 + 16 + half * 8 + r] = p1[r];
            }
        }
        __syncthreads();

        // deterministic tree-free reduction of per-wave partials (no fp atomics)
        if (tid < 32) {
            float sacc = 0.0f;
            #pragma unroll
            for (int w = 0; w < NTILES; ++w) sacc += spart[w * 32 + tid];
            scores[tid] = sacc;
        }
        __syncthreads();

        // ---- phase C: softmax over 26 entries + weighted V mix (register-resident V) ----
        float mx = scores[0];
        #pragma unroll
        for (int j = 1; j < ROWS; ++j) mx = fmaxf(mx, scores[j]);
        float ssum = 0.0f;
        #pragma unroll
        for (int j = 0; j < ROWS; ++j) ssum += __expf(scores[j] - mx);
        const float inv = 1.0f / ssum;

        float acc = 0.0f;
        #pragma unroll
        for (int r = 0; r < 8; ++r) {
            int j0 = r + half * 8;            // rows 0..15: always valid
            acc += __expf(scores[j0] - mx) * accv0[r];
            int j1 = 16 + r + half * 8;       // rows 16..31: valid while < 26
            if (j1 < ROWS) acc += __expf(scores[j1] - mx) * accv1[r];
        }
        acc *= inv;
        acc += __shfl_xor(acc, 16, 32);       // combine the two lane-halves (same column)
        if (half == 0) out[(size_t)nd * DIM + ncol] = acc;
        __syncthreads();  // protect scores/spart before next node's writes
    }
}

extern "C" void kernel_launch(void* const* d_in, const int* in_sizes, int n_in,
                              void* d_out, int out_size, void* d_ws, size_t ws_size,
                              hipStream_t stream)
{
    const float* table  = (const float*)d_in[0];
    const int*   node   = (const int*)  d_in[1];
    const int*   neighs = (const int*)  d_in[2];
    const float* Wq     = (const float*)d_in[3];
    const float* bq     = (const float*)d_in[4];
    const float* Wk     = (const float*)d_in[5];
    const float* bk     = (const float*)d_in[6];
    const float* Wv     = (const float*)d_in[7];
    const float* bv     = (const float*)d_in[8];
    float* out = (float*)d_out;

    const int N = in_sizes[1];           // number of nodes
    int blocks = N < 1024 ? N : 1024;    // ~1 WG per WGP (LDS-bound), loop nodes inside

    attn_agg_kernel<<<dim3(blocks), dim3(WGSIZE), 0, stream>>>(
        table, node, neighs, Wq, bq, Wk, bk, Wv, bv, out, N);
}